// MarcoPoloModel_84911503442063
// MI455X (gfx1250) — compile-verified
//
#include <hip/hip_runtime.h>

typedef __attribute__((ext_vector_type(2))) float v2f;
typedef __attribute__((ext_vector_type(8))) float v8f;

#define N_CELLS 16384
#define G_GENES 1024
#define C_COV   5

// One wave = one 16-row slab. Block = 256 thr = 8 waves = 128 rows. Grid = 128.
__global__ __launch_bounds__(256) void marcopolo_fused(
    const float* __restrict__ Y, const float* __restrict__ X,
    const float* __restrict__ s, const float* __restrict__ delta_log,
    const float* __restrict__ beta, const float* __restrict__ mask,
    float* __restrict__ lse_out)
{
    __shared__ float lds_d[2][G_GENES];    // d = exp(delta_log)*mask
    __shared__ float lds_ed[2][G_GENES];   // exp(d)
    __shared__ float lds_lg[32];           // log(k!) table for Y in [0,19]
    __shared__ float lds_y[8][2][256];     // per-wave double-buffered 16x16 Y tile

    const int tid = threadIdx.x;
    for (int i = tid; i < 2 * G_GENES; i += 256) {
        float dv = __expf(delta_log[i]) * mask[i];
        int k = i >> 10, g = i & (G_GENES - 1);
        lds_d[k][g]  = dv;
        lds_ed[k][g] = __expf(dv);
    }
    if (tid < 32) {
        float a = 0.f;
        for (int j = 2; j <= tid; ++j) a += __logf((float)j);
        lds_lg[tid] = a; // lgamma(tid+1) = log(tid!)
    }
    __syncthreads();

    const int lane = tid & 31;
    const int wave = tid >> 5;
    const int half = lane >> 4;   // 0: lanes 0-15, 1: lanes 16-31
    const int col  = lane & 15;
    const int rowbase = (blockIdx.x * 8 + wave) * 16;
    const float c4 = (half == 0) ? 1.f : 0.f;  // K=4 pad mask (branch-free)

    // --- A operand (X tile, 16x4 f32 layout): loop-invariant ---
    // A layout: all lanes M = lane&15; VGPR0 holds K=0 (half 0) / K=2 (half 1), VGPR1 K=1/K=3
    const int arow = rowbase + col;
    const int c0   = half * 2;
    v2f a1, a2;
    a1.x = X[arow * C_COV + c0];
    a1.y = X[arow * C_COV + c0 + 1];
    a2.x = X[arow * C_COV + 4] * c4;  // K=4 real, K=5..7 zero pad
    a2.y = 0.f;

    float ls[8];
#pragma unroll
    for (int j = 0; j < 8; ++j)
        ls[j] = __logf(s[rowbase + j + 8 * half]);  // matches C/D accumulator row map

    // Async-copy slot assignment: 64 x b128 slots cover 16 rows x 16 genes
    const int jA = lane >> 2,        qA = lane & 3;         // slots 0..31
    const int jB = (lane + 32) >> 2, qB = lane & 3;         // slots 32..63
    const unsigned ybase = (unsigned)(uintptr_t)&lds_y[wave][0][0];
    const unsigned ldsA  = (unsigned)((jA * 16 + qA * 4) * 4);
    const unsigned ldsB  = (unsigned)((jB * 16 + qB * 4) * 4);

    float p0[8], p1[8];
#pragma unroll
    for (int j = 0; j < 8; ++j) p0[j] = p1[j] = 0.f;

    // Prologue: stage chunk 0 into buffer 0
    {
        const float* gA = Y + (long)(rowbase + jA) * G_GENES + qA * 4;
        const float* gB = Y + (long)(rowbase + jB) * G_GENES + qB * 4;
        asm volatile("global_load_async_to_lds_b128 %0, %1, off"
                     :: "v"(ybase + ldsA), "v"(gA) : "memory");
        asm volatile("global_load_async_to_lds_b128 %0, %1, off"
                     :: "v"(ybase + ldsB), "v"(gB) : "memory");
    }

    for (int chunk = 0; chunk < G_GENES / 16; ++chunk) {
        const int gbase = chunk << 4;
        const int buf   = chunk & 1;
        const int gcol  = gbase + col;

        if (chunk + 1 < G_GENES / 16) {  // uniform branch
            const int gn = (chunk + 1) << 4;
            const unsigned bofs = ybase + (unsigned)((buf ^ 1) * 1024);
            const float* gA = Y + (long)(rowbase + jA) * G_GENES + gn + qA * 4;
            const float* gB = Y + (long)(rowbase + jB) * G_GENES + gn + qB * 4;
            asm volatile("global_load_async_to_lds_b128 %0, %1, off"
                         :: "v"(bofs + ldsA), "v"(gA) : "memory");
            asm volatile("global_load_async_to_lds_b128 %0, %1, off"
                         :: "v"(bofs + ldsB), "v"(gB) : "memory");
            asm volatile("s_wait_asynccnt 0x2" ::: "memory"); // chunk c staged
        } else {
            asm volatile("s_wait_asynccnt 0x0" ::: "memory");
        }

        // --- B operand (beta tile, 4x16 f32): VGPR0 = K=0/K=2 by half, VGPR1 = K=1/K=3 ---
        v2f b1, b2;
        b1.x = beta[c0 * G_GENES + gcol];
        b1.y = beta[(c0 + 1) * G_GENES + gcol];
        b2.x = beta[4 * G_GENES + gcol] * c4;  // unconditional load, masked pad
        b2.y = 0.f;

        // base tile = X(16x5) @ beta(5x16), padded K=8 via two 16x16x4 WMMAs
        v8f acc = {0.f, 0.f, 0.f, 0.f, 0.f, 0.f, 0.f, 0.f};
        acc = __builtin_amdgcn_wmma_f32_16x16x4_f32(false, a1, false, b1,
                                                    (short)0, acc, false, false);
        acc = __builtin_amdgcn_wmma_f32_16x16x4_f32(false, a2, false, b2,
                                                    (short)0, acc, false, false);

        const float dv0 = lds_d[0][gcol],  dv1 = lds_d[1][gcol];
        const float ed0 = lds_ed[0][gcol], ed1 = lds_ed[1][gcol];
        const float* ytile = &lds_y[wave][buf][0];

#pragma unroll
        for (int j = 0; j < 8; ++j) {
            float b = acc[j] + ls[j];                  // eta base (row j+8*half, gene gcol)
            float e = __expf(b);                       // exp(base); mu_k = e*exp(d_k)
            float y = ytile[(j + 8 * half) * 16 + col];
            float t = fmaf(y, b, -lds_lg[(int)y]);     // y*eta_base - lgamma(y+1)
            p0[j] += fmaf(y, dv0, t) - e * ed0;        // + y*d0 - mu0
            p1[j] += fmaf(y, dv1, t) - e * ed1;        // + y*d1 - mu1
        }
    }

    // Reduce over the 16 lanes of each half (gene columns). xor masks 1..8 stay in-half.
#pragma unroll
    for (int j = 0; j < 8; ++j) {
        for (int off = 1; off < 16; off <<= 1) {
            p0[j] += __shfl_xor(p0[j], off, 32);
            p1[j] += __shfl_xor(p1[j], off, 32);
        }
    }

    if (col < 8) {
        const int j   = col;
        const int row = rowbase + j + 8 * half;
        float r0 = p0[j], r1 = p1[j];
        float m  = fmaxf(r0, r1);
        float lse = m + __logf(__expf(r0 - m) + __expf(r1 - m));
        lse_out[row] = lse;
    }
}

// Deterministic final reduction: fixed strided order + fixed tree.
__global__ __launch_bounds__(256) void marcopolo_reduce(
    const float* __restrict__ lse, float* __restrict__ out)
{
    __shared__ float sm[256];
    float acc = 0.f;
    for (int i = threadIdx.x; i < N_CELLS; i += 256) acc += lse[i];
    sm[threadIdx.x] = acc;
    __syncthreads();
    for (int stride = 128; stride > 0; stride >>= 1) {
        if (threadIdx.x < stride) sm[threadIdx.x] += sm[threadIdx.x + stride];
        __syncthreads();
    }
    if (threadIdx.x == 0) out[0] = sm[0];
}

extern "C" void kernel_launch(void* const* d_in, const int* in_sizes, int n_in,
                              void* d_out, int out_size, void* d_ws, size_t ws_size,
                              hipStream_t stream) {
    (void)in_sizes; (void)n_in; (void)out_size; (void)ws_size;
    const float* Y         = (const float*)d_in[0];
    const float* X         = (const float*)d_in[1];
    const float* s         = (const float*)d_in[2];
    const float* delta_log = (const float*)d_in[3];
    const float* beta      = (const float*)d_in[4];
    const float* mask      = (const float*)d_in[5];
    float* out = (float*)d_out;
    float* ws  = (float*)d_ws;   // N_CELLS floats of per-row logsumexp

    marcopolo_fused<<<N_CELLS / 128, 256, 0, stream>>>(Y, X, s, delta_log, beta, mask, ws);
    marcopolo_reduce<<<1, 256, 0, stream>>>(ws, out);
}